// RandAugment_cuda_67662914781509
// MI455X (gfx1250) — compile-verified
//
#include <hip/hip_runtime.h>
#include <stdint.h>

// ---------------- problem constants ----------------
#define B_      256
#define C_      3
#define H_      224
#define W_      224
#define PLANE_  (H_ * W_)          // 50176
#define TH_     16                 // output rows per tile
#define NTILE_H (H_ / TH_)         // 14
#define SPLITH  8                  // row-splits per plane in pass 1
#define RPS     (H_ / SPLITH)      // 28 rows per pass-1 block

// M_MAG = 9 : v_blend = 9/30*1.8+0.1 = 0.64 ; v_sol = 0.3
constexpr float VB    = 0.64f;
constexpr float OMVB  = 1.0f - 0.64f;       // 0.36
constexpr float VSOL  = 0.3f;
constexpr float INV13 = 1.0f / 13.0f;
// mean of mass(h,w) over the 224x224 plane:
//   (sum(nvalid) + 4*HW)/13/HW = (448900+200704)/13/50176
constexpr float MEANMASS = 649604.0f / 652288.0f;

typedef unsigned int v4u __attribute__((ext_vector_type(4)));
typedef int          v4i __attribute__((ext_vector_type(4)));
typedef int          v8i __attribute__((ext_vector_type(8)));

// Require the TDM builtin in the DEVICE pass only (host pass lacks all
// __builtin_amdgcn_* and must not trip this).
#if defined(__HIP_DEVICE_COMPILE__)
#if !__has_builtin(__builtin_amdgcn_tensor_load_to_lds)
#error "tensor_load_to_lds builtin not available on this device toolchain"
#endif
#endif

// ---------------- pass 1: weighted reduction of x -> per-block partials ----
// Weight = VB + OMVB*mass(h,w); mass==1 (weight==1) for all interior pixels,
// so the hot path is a plain sum. Only boundary rows/cols take the slow path.
__global__ __launch_bounds__(256)
void k_gm_partial(const float* __restrict__ x,
                  const float* __restrict__ std_,
                  float* __restrict__ part)
{
    const int blk   = blockIdx.x;
    const int sh    = blk & (SPLITH - 1);
    const int plane = blk >> 3;                 // b*3 + c
    const int c     = plane % 3;
    const int h0    = sh * RPS;

    const float4* __restrict__ px =
        (const float4*)(x + (size_t)plane * PLANE_ + (size_t)h0 * W_);

    float acc = 0.f;
    const int n4 = RPS * (W_ / 4);              // 1568 float4s
    for (int i = threadIdx.x; i < n4; i += blockDim.x) {
        const int r  = i / (W_ / 4);
        const int w0 = (i - r * (W_ / 4)) * 4;
        const int h  = h0 + r;
        const float4 v = px[i];
        const float s4 = (v.x + v.y) + (v.z + v.w);
        if (h > 0 && h < H_ - 1 && w0 > 0 && w0 < W_ - 4) {
            acc += s4;                          // interior: weight == 1
        } else {
            const float hn = (h > 0 && h < H_ - 1) ? 3.f : 2.f;
            const float* e = &v.x;
#pragma unroll
            for (int j = 0; j < 4; ++j) {
                const int   w   = w0 + j;
                const float wn  = (w > 0 && w < W_ - 1) ? 3.f : 2.f;
                const float wgt = VB + OMVB * (hn * wn + 4.f) * INV13;
                acc += e[j] * wgt;
            }
        }
    }
    // fold gray weight and std[c] into this block's partial
    const float grayw = (c == 0) ? 0.299f : (c == 1 ? 0.587f : 0.114f);
    acc *= grayw * std_[c];

    __shared__ float red[256];
    red[threadIdx.x] = acc;
    __syncthreads();
#pragma unroll
    for (int s = 128; s > 0; s >>= 1) {
        if ((int)threadIdx.x < s) red[threadIdx.x] += red[threadIdx.x + s];
        __syncthreads();
    }
    if (threadIdx.x == 0) part[blk] = red[0];
}

// ---------------- pass 1b: finalize gm[b] (deterministic fixed order) ------
__global__ __launch_bounds__(256)
void k_gm_final(const float* __restrict__ part,
                const float* __restrict__ mean_,
                float* __restrict__ gm)
{
    const int b = threadIdx.x;                  // one block of 256 threads
    float s = 0.f;
#pragma unroll
    for (int i = 0; i < C_ * SPLITH; ++i)
        s += part[b * C_ * SPLITH + i];
    const float cterm =
        (0.299f * mean_[0] + 0.587f * mean_[1] + 0.114f * mean_[2]) *
        (VB + OMVB * MEANMASS);
    gm[b] = s * (1.f / (float)PLANE_) + cterm;
}

// ---------------- pass 2: TDM-fed fused blur + blend + solarize + renorm ---
__global__ __launch_bounds__(256)
void k_apply(const float* __restrict__ x,
             const float* __restrict__ mean_,
             const float* __restrict__ std_,
             const float* __restrict__ gm,
             float* __restrict__ out)
{
    __shared__ float tile[(TH_ + 2) * W_];      // 18 rows x 224 cols, packed

    const int blk   = blockIdx.x;
    const int th    = blk % NTILE_H;
    const int plane = blk / NTILE_H;            // b*3 + c
    const int b     = plane / C_;
    const int c     = plane % C_;
    const int h0    = th * TH_;

    // zero top/bottom halo rows only at image boundary tiles
    if (h0 == 0)
        for (int i = threadIdx.x; i < W_; i += blockDim.x) tile[i] = 0.f;
    if (h0 + TH_ == H_)
        for (int i = threadIdx.x; i < W_; i += blockDim.x)
            tile[(TH_ + 1) * W_ + i] = 0.f;

    const int grow0 = (h0 == 0) ? 0 : (h0 - 1);
    const int grow1 = (h0 + TH_ + 1 < H_) ? (h0 + TH_ + 1) : H_;
    const int nelem = (grow1 - grow0) * W_;     // <= 4032, one contiguous run

#if defined(__HIP_DEVICE_COMPILE__)
    if (threadIdx.x < 32) {                     // wave 0 issues the TDM DMA
        const uint64_t ga =
            (uint64_t)(uintptr_t)(x + (size_t)plane * PLANE_ +
                                  (size_t)grow0 * W_);
        const uint32_t lds =
            (uint32_t)(uintptr_t)(void*)tile + (uint32_t)((h0 == 0) ? W_ * 4 : 0);

        // D# group 0: count=1 | lds_addr | global_addr | type=2
        v4u g0;
        g0[0] = (unsigned)__builtin_amdgcn_readfirstlane((int)1u);
        g0[1] = (unsigned)__builtin_amdgcn_readfirstlane((int)lds);
        g0[2] = (unsigned)__builtin_amdgcn_readfirstlane((int)(uint32_t)ga);
        g0[3] = (unsigned)__builtin_amdgcn_readfirstlane(
                    (int)((uint32_t)(ga >> 32) | (2u << 30)));

        // D# group 1: data_size=4B, tensor_dim0 = tile_dim0 = nelem (1-D copy)
        v8i g1;
        g1[0] = __builtin_amdgcn_readfirstlane((int)(2u << 16));               // data_size=2 (4B)
        g1[1] = __builtin_amdgcn_readfirstlane((int)((nelem & 0xFFFF) << 16)); // tensor_dim0 lo16
        g1[2] = __builtin_amdgcn_readfirstlane((int)(((nelem >> 16) & 0xFFFF) |
                                                     (1 << 16)));              // dim0 hi | tensor_dim1=1
        g1[3] = __builtin_amdgcn_readfirstlane((int)(nelem << 16));            // tile_dim0 = nelem
        g1[4] = __builtin_amdgcn_readfirstlane(1);                             // tile_dim1=1
        g1[5] = __builtin_amdgcn_readfirstlane(nelem);                         // dim0 stride (unused)
        g1[6] = 0;
        g1[7] = 0;

        v4i z4 = (v4i)0;
#if defined(__clang_major__) && (__clang_major__ >= 23)
        v8i z8 = (v8i)0;
        __builtin_amdgcn_tensor_load_to_lds(g0, g1, z4, z4, z8, 0);
#else
        __builtin_amdgcn_tensor_load_to_lds(g0, g1, z4, z4, 0);
#endif
        __builtin_amdgcn_s_wait_tensorcnt(0);
    }
#else
    // Host-compilation pass only: parse-only body, never executed or
    // device-codegen'd (device pass hard-requires the TDM builtin above).
    {
        const float* src = x + (size_t)plane * PLANE_ + (size_t)grow0 * W_;
        float* dst = tile + ((h0 == 0) ? W_ : 0);
        for (int i = threadIdx.x; i < nelem; i += blockDim.x) dst[i] = src[i];
    }
#endif
    __syncthreads();

    const float m    = mean_[c];
    const float s    = std_[c];
    const float invs = 1.f / s;
    const float g    = gm[b];
    float* __restrict__ op = out + (size_t)plane * PLANE_ + (size_t)h0 * W_;

    // 16*224/256 = 14 pixels per thread, coalesced stores
    for (int p = threadIdx.x; p < TH_ * W_; p += blockDim.x) {
        const int rl = p / W_;
        const int cc = p - rl * W_;
        const int h  = h0 + rl;
        const float* r0 = tile + rl * W_;       // image row h-1
        const float* r1 = r0 + W_;              // image row h
        const float* r2 = r1 + W_;              // image row h+1
        const bool cl = (cc > 0), cr = (cc < W_ - 1);

        const float xc = r1[cc];
        const float a0 = (cl ? r0[cc - 1] : 0.f) + r0[cc] + (cr ? r0[cc + 1] : 0.f);
        const float a1 = (cl ? r1[cc - 1] : 0.f) + xc     + (cr ? r1[cc + 1] : 0.f);
        const float a2 = (cl ? r2[cc - 1] : 0.f) + r2[cc] + (cr ? r2[cc + 1] : 0.f);
        const float blurx = (a0 + a1 + a2 + 4.f * xc) * INV13;  // (box + 4*center)/13

        const float hn    = (h > 0 && h < H_ - 1) ? 3.f : 2.f;
        const float wn    = (cl && cr) ? 3.f : 2.f;
        const float massf = (hn * wn + 4.f) * INV13;

        float y = s * (xc * VB + blurx * OMVB) + m * (VB + OMVB * massf); // y3
        y = y * VB + g * OMVB;                                            // + gray mean
        y = y * VB;
        y = (y < VSOL) ? y : (1.f - y);                                   // solarize
        op[p] = (y - m) * invs;                                           // renormalize
    }
}

// ---------------- host launcher -------------------------------------------
extern "C" void kernel_launch(void* const* d_in, const int* in_sizes, int n_in,
                              void* d_out, int out_size, void* d_ws, size_t ws_size,
                              hipStream_t stream)
{
    const float* x     = (const float*)d_in[0];
    const float* mean_ = (const float*)d_in[1];
    const float* std_  = (const float*)d_in[2];
    float*       out   = (float*)d_out;

    float* part = (float*)d_ws;        // B*C*SPLITH = 6144 floats of partials
    float* gm   = part + 8192;         // 256 floats of per-image gray means

    hipLaunchKernelGGL(k_gm_partial, dim3(B_ * C_ * SPLITH), dim3(256), 0, stream,
                       x, std_, part);
    hipLaunchKernelGGL(k_gm_final, dim3(1), dim3(256), 0, stream,
                       part, mean_, gm);
    hipLaunchKernelGGL(k_apply, dim3(B_ * C_ * NTILE_H), dim3(256), 0, stream,
                       x, mean_, std_, gm, out);
}